// CRFModel_80487687127195
// MI455X (gfx1250) — compile-verified
//
#include <hip/hip_runtime.h>
#include <hip/hip_bf16.h>

typedef __attribute__((ext_vector_type(2))) float v2f;
typedef __attribute__((ext_vector_type(8))) float v8f;

// Problem constants (from reference)
static constexpr int kVocab = 32000;
static constexpr int kBeam  = 64;
static constexpr int kRank  = 32;
static constexpr int kB     = 4;
static constexpr int kT     = 512;
static constexpr int kBT    = kB * kT;   // 2048
static constexpr int kE     = 1024;

// GEMM tiling
static constexpr int BM = 64;
static constexpr int BN = 128;
static constexpr int KC = 32;
static constexpr int KP = KC + 1;        // pad: gcd(33,64)=1 -> conflict-free column reads

// ---------------------------------------------------------------------------
// Async global->LDS copy (CDNA5 GLOBAL_LOAD_ASYNC_TO_LDS_B32, ASYNCcnt),
// with synchronous fallback if the builtin is unavailable on this toolchain.
// Builtin signature (from hipcc diagnostic): (AS1 int*, AS3 int*, Imm, Imm).
// ---------------------------------------------------------------------------
#if defined(__HIP_DEVICE_COMPILE__) && __has_builtin(__builtin_amdgcn_global_load_async_to_lds_b32)
#define HAS_ASYNC_LDS 1
#else
#define HAS_ASYNC_LDS 0
#endif

__device__ __forceinline__ void cp_b32_to_lds(float* lds_dst, const float* g_src) {
#if HAS_ASYNC_LDS
  auto g1 = reinterpret_cast<__attribute__((address_space(1))) int*>(
      reinterpret_cast<uintptr_t>(g_src));
  auto l3 = reinterpret_cast<__attribute__((address_space(3))) int*>(
      reinterpret_cast<uintptr_t>(lds_dst));
  __builtin_amdgcn_global_load_async_to_lds_b32(g1, l3, 0, 0);
#else
  *lds_dst = *g_src;
#endif
}

__device__ __forceinline__ void wait_async_lds() {
#if HAS_ASYNC_LDS
#if __has_builtin(__builtin_amdgcn_s_wait_asynccnt)
  __builtin_amdgcn_s_wait_asynccnt(0);
#else
  asm volatile("s_wait_asynccnt 0" ::: "memory");
#endif
#endif
}

// ---------------------------------------------------------------------------
// Kernel A: logits[m, v] = sum_e A[m, e] * W[v, e] + bias[v]
// 256 threads = 8 waves; wave grid 4(M) x 2(N); each wave: 16x64 = 4 WMMA tiles.
// Double-buffered LDS, async global->LDS staging overlapped with WMMA math.
// ---------------------------------------------------------------------------
__global__ __launch_bounds__(256) void gemm_logits(
    const float* __restrict__ A, const float* __restrict__ W,
    const float* __restrict__ bias, float* __restrict__ out)
{
  __shared__ float As[2][BM][KP];   // 2 x 64 x 33
  __shared__ float Bs[2][BN][KP];   // 2 x 128 x 33

  const int tid  = threadIdx.x;
  const int lane = tid & 31;
  const int wave = tid >> 5;            // 0..7
  const int wm   = (wave >> 1) * 16;    // 0,16,32,48
  const int wn   = (wave & 1) * 64;     // 0,64
  const int mrow = lane & 15;
  const int khalf = (lane >> 4) * 2;    // 0 or 2 (A/B fragment K pair)

  const int m0 = blockIdx.y * BM;
  const int n0 = blockIdx.x * BN;

  // accumulators initialized with bias (bias depends on column only)
  v8f acc[4];
  for (int nt = 0; nt < 4; ++nt) {
    float bv = bias[n0 + wn + nt * 16 + mrow];
    v8f c; c[0]=bv; c[1]=bv; c[2]=bv; c[3]=bv; c[4]=bv; c[5]=bv; c[6]=bv; c[7]=bv;
    acc[nt] = c;
  }

  // Tile staging: element e of a (rows x 32) tile -> row = e>>5, col = e&31.
  // Consecutive lanes cover one full 128B row per instruction (coalesced),
  // LDS destination is contiguous within the padded row.
  auto issue_chunk = [&](int buf, int k0) {
    for (int j = 0; j < BM * KC / 256; ++j) {          // A: 2048 elems -> 8 per thread
      const int e = tid + j * 256;
      const int r = e >> 5, c = e & 31;
      cp_b32_to_lds(&As[buf][r][c], A + (size_t)(m0 + r) * kE + k0 + c);
    }
    for (int j = 0; j < BN * KC / 256; ++j) {          // B: 4096 elems -> 16 per thread
      const int e = tid + j * 256;
      const int r = e >> 5, c = e & 31;
      cp_b32_to_lds(&Bs[buf][r][c], W + (size_t)(n0 + r) * kE + k0 + c);
    }
  };

  auto compute_chunk = [&](int buf) {
    for (int kk = 0; kk < KC; kk += 4) {
      // A fragment (16x4): lane m holds K = khalf, khalf+1 of row m
      v2f a;
      a.x = As[buf][wm + mrow][kk + khalf];
      a.y = As[buf][wm + mrow][kk + khalf + 1];
      for (int nt = 0; nt < 4; ++nt) {
        // B fragment (4x16): lane n holds K = khalf, khalf+1 of column n
        v2f bf;
        bf.x = Bs[buf][wn + nt * 16 + mrow][kk + khalf];
        bf.y = Bs[buf][wn + nt * 16 + mrow][kk + khalf + 1];
        acc[nt] = __builtin_amdgcn_wmma_f32_16x16x4_f32(
            false, a, false, bf, (short)0, acc[nt], false, false);
      }
    }
  };

  constexpr int kChunks = kE / KC;   // 32
  issue_chunk(0, 0);
  wait_async_lds();
  __syncthreads();

  for (int ch = 0; ch < kChunks; ++ch) {
    const int cur = ch & 1;
    if (ch + 1 < kChunks) issue_chunk(cur ^ 1, (ch + 1) * KC);  // overlap with math
    compute_chunk(cur);
    wait_async_lds();      // next chunk's async copies done (own wave)
    __syncthreads();       // ...and visible to / reusable by all waves
  }

  // C/D layout: VGPR i -> row (i + 8*(lane>=16)), col (lane&15)
  const int rbase = m0 + wm + ((lane >> 4) << 3);
  for (int nt = 0; nt < 4; ++nt) {
    const int col = n0 + wn + nt * 16 + mrow;
    for (int i = 0; i < 8; ++i) {
      out[(size_t)(rbase + i) * kVocab + col] = acc[nt][i];
    }
  }
}

// ---------------------------------------------------------------------------
// Kernel B: exact top-64 indices of forced logits per (b,t).
// Key = ordered(value) << 32 | (0xFFFFFFFF - idx)  => value desc, index asc.
// ---------------------------------------------------------------------------
__global__ __launch_bounds__(256) void topk_beam(
    const float* __restrict__ logits, const int* __restrict__ target,
    int* __restrict__ beams)
{
  const int bt = blockIdx.x;
  const float* row = logits + (size_t)bt * kVocab;
  const int tgt = target[bt];
  const int tid = threadIdx.x;

  __shared__ unsigned long long red[256];

  unsigned long long bound = ~0ull;   // strict exclusive upper bound
  for (int j = 0; j < kBeam; ++j) {
    unsigned long long best = 0ull;
    for (int i = tid; i < kVocab; i += 256) {
      float v = (i == tgt) ? __builtin_inff() : row[i];
      unsigned u = __float_as_uint(v);
      u = (u & 0x80000000u) ? ~u : (u | 0x80000000u);
      unsigned long long key =
          ((unsigned long long)u << 32) | (unsigned)(0xFFFFFFFFu - (unsigned)i);
      if (key < bound && key > best) best = key;
    }
    red[tid] = best;
    __syncthreads();
    for (int s = 128; s > 0; s >>= 1) {
      if (tid < s) {
        unsigned long long o = red[tid + s];
        if (o > red[tid]) red[tid] = o;
      }
      __syncthreads();
    }
    const unsigned long long win = red[0];
    __syncthreads();
    if (tid == 0)
      beams[(size_t)bt * kBeam + j] = (int)(0xFFFFFFFFu - (unsigned)(win & 0xFFFFFFFFu));
    bound = win;
  }
}

// ---------------------------------------------------------------------------
// Kernel C: per-batch numerator + beam forward scan -> loss = denom - num
// One 64-thread block per batch; thread l owns beam slot l.
// ---------------------------------------------------------------------------
__global__ __launch_bounds__(64) void crf_scan(
    const float* __restrict__ logits, const int* __restrict__ target,
    const float* __restrict__ E1, const float* __restrict__ E2,
    const int* __restrict__ beams, float* __restrict__ loss_out)
{
  const int b = blockIdx.x;
  const int tid = threadIdx.x;   // 0..63

  __shared__ float e1s[kBeam][kRank];
  __shared__ float sc[kBeam];
  __shared__ float redf[64];

  // ---- numerator: sum_t mask[t] * (emit[t] + (t>0 ? trans[t] : 0)) ----
  float num = 0.f;
  for (int t = tid; t < kT; t += 64) {
    const int tg = target[b * kT + t];
    if (tg != 0) {
      float s = logits[(size_t)(b * kT + t) * kVocab + tg];
      if (t > 0) {
        const int tp = target[b * kT + t - 1];
        float d = 0.f;
        for (int r = 0; r < kRank; ++r)
          d += E1[(size_t)tp * kRank + r] * E2[(size_t)tg * kRank + r];
        s += d;
      }
      num += s;
    }
  }
  redf[tid] = num;
  __syncthreads();
  for (int s = 32; s > 0; s >>= 1) {
    if (tid < s) redf[tid] += redf[tid + s];
    __syncthreads();
  }
  const float numerator = redf[0];
  __syncthreads();

  // ---- forward scan over beam ----
  int cur = beams[(size_t)(b * kT + 0) * kBeam + tid];
  float score = logits[(size_t)(b * kT + 0) * kVocab + cur];

  for (int t = 1; t < kT; ++t) {
    // previous-step beam: stage score vector and E1 rows in LDS
    const int pk = beams[(size_t)(b * kT + t - 1) * kBeam + tid];
    sc[tid] = score;
    for (int r = 0; r < kRank; ++r) e1s[tid][r] = E1[(size_t)pk * kRank + r];
    __syncthreads();

    const int cl = beams[(size_t)(b * kT + t) * kBeam + tid];
    float e2r[kRank];
    for (int r = 0; r < kRank; ++r) e2r[r] = E2[(size_t)cl * kRank + r];

    // online logsumexp over k of (score[k] + <E1[prev_k], E2[cur_l]>)
    float m = -__builtin_inff(), ssum = 0.f;
    for (int k = 0; k < kBeam; ++k) {
      float d = 0.f;
      for (int r = 0; r < kRank; ++r) d += e1s[k][r] * e2r[r];
      const float x = sc[k] + d;
      if (x > m) { ssum = ssum * __expf(m - x) + 1.f; m = x; }
      else       { ssum += __expf(x - m); }
    }
    const float nxt = m + __logf(ssum)
                    + logits[(size_t)(b * kT + t) * kVocab + cl];
    if (target[b * kT + t] != 0) score = nxt;
    __syncthreads();   // protect sc/e1s before next iteration overwrites
  }

  // ---- denominator = logsumexp over beam slots ----
  redf[tid] = score;
  __syncthreads();
  for (int s = 32; s > 0; s >>= 1) {
    if (tid < s) redf[tid] = fmaxf(redf[tid], redf[tid + s]);
    __syncthreads();
  }
  const float gmax = redf[0];
  __syncthreads();
  redf[tid] = __expf(score - gmax);
  __syncthreads();
  for (int s = 32; s > 0; s >>= 1) {
    if (tid < s) redf[tid] += redf[tid + s];
    __syncthreads();
  }
  if (tid == 0) {
    const float denom = gmax + __logf(redf[0]);
    loss_out[b] = denom - numerator;
  }
}

// ---------------------------------------------------------------------------
extern "C" void kernel_launch(void* const* d_in, const int* in_sizes, int n_in,
                              void* d_out, int out_size, void* d_ws, size_t ws_size,
                              hipStream_t stream) {
  const float* modelout = (const float*)d_in[0];
  const float* W        = (const float*)d_in[1];
  const float* bias     = (const float*)d_in[2];
  const float* E1       = (const float*)d_in[3];
  const float* E2       = (const float*)d_in[4];
  const int*   target   = (const int*)d_in[5];

  float* out  = (float*)d_out;                 // logits | 4 losses
  int* beams  = (int*)d_ws;                    // 2048 * 64 ints = 512 KB

  dim3 gg(kVocab / BN, kBT / BM);              // 250 x 32
  gemm_logits<<<gg, 256, 0, stream>>>(modelout, W, bias, out);
  topk_beam<<<kBT, 256, 0, stream>>>(out, target, beams);
  crf_scan<<<kB, 64, 0, stream>>>(out, target, E1, E2, beams,
                                  out + (size_t)kBT * kVocab);
}